// LSTM_pred_85641647882937
// MI455X (gfx1250) — compile-verified
//
#include <hip/hip_runtime.h>

typedef float v2f __attribute__((ext_vector_type(2)));
typedef float v8f __attribute__((ext_vector_type(8)));

#define SEQ    96
#define TIN    13
#define HID    10
#define OUTN   9
#define PITCH  20   // LDS row pitch (floats): conflict-free for store & gather patterns

#define WMMA_F32(a, b, c) \
  __builtin_amdgcn_wmma_f32_16x16x4_f32(false, (a), false, (b), (short)0, (c), false, false)

__device__ __forceinline__ void lds_fence() {
  asm volatile("s_wait_dscnt 0" ::: "memory");
}
__device__ __forceinline__ float sigf(float x) {
  return __builtin_amdgcn_rcpf(1.0f + __expf(-x));
}
__device__ __forceinline__ float tanhfast(float x) {
  // tanh(x) = 1 - 2/(exp(2x)+1); saturates correctly at +/-1
  return 1.0f - 2.0f * __builtin_amdgcn_rcpf(__expf(2.0f * x) + 1.0f);
}

__global__ __launch_bounds__(256) void lstm_encdec_kernel(
    const float* __restrict__ x,    const float* __restrict__ ta,
    const float* __restrict__ Wih1, const float* __restrict__ Whh1,
    const float* __restrict__ bih1, const float* __restrict__ bhh1,
    const float* __restrict__ Wih2, const float* __restrict__ Whh2,
    const float* __restrict__ bih2, const float* __restrict__ bhh2,
    const float* __restrict__ Wl,   const float* __restrict__ bl,
    float* __restrict__ outp)
{
  __shared__ float smem[8 * 3 * 16 * PITCH];   // 8 waves x {h1,h2,out} x [16][PITCH]

  const int tid  = threadIdx.x;
  const int wid  = tid >> 5;
  const int lane = tid & 31;
  const int nn   = lane & 15;   // D-layout: column (hidden/gate unit); A-layout: batch row
  const int kh   = lane >> 4;   // K half-select for A/B operands
  const int b0   = (blockIdx.x * 8 + wid) * 16;

  float* ldsH1  = &smem[wid * 3 * 16 * PITCH];
  float* ldsH2  = ldsH1 + 16 * PITCH;
  float* ldsOut = ldsH2 + 16 * PITCH;

  // ---------------- preload weights into WMMA B-layout registers ----------------
  // B 4x16 layout: lane = N (+16 selects K pair), component j -> K = 4*kk + 2*kh + j
  v2f w1[4][6];   // layer1: K = [x(13) | h1(10) | pad] -> 6 K-steps, 4 gate tiles
  v2f w2[4][5];   // layer2: K = [h1(10) | h2(10)]      -> 5 K-steps
  v2f wl[3];      // out:    K = [h2(10) | pad]         -> 3 K-steps
  float b1[4], b2[4], blv;

  #pragma unroll
  for (int g = 0; g < 4; ++g) {
    #pragma unroll
    for (int kk = 0; kk < 6; ++kk) {
      #pragma unroll
      for (int j = 0; j < 2; ++j) {
        int k = 4 * kk + 2 * kh + j;
        float v = 0.0f;
        if (nn < HID) {
          int row = g * HID + nn;
          if (k < TIN)            v = Wih1[row * TIN + k];
          else if (k < TIN + HID) v = Whh1[row * HID + (k - TIN)];
        }
        w1[g][kk][j] = v;
      }
    }
    #pragma unroll
    for (int kk = 0; kk < 5; ++kk) {
      #pragma unroll
      for (int j = 0; j < 2; ++j) {
        int k = 4 * kk + 2 * kh + j;
        float v = 0.0f;
        if (nn < HID) {
          int row = g * HID + nn;
          if (k < HID)          v = Wih2[row * HID + k];
          else if (k < 2 * HID) v = Whh2[row * HID + (k - HID)];
        }
        w2[g][kk][j] = v;
      }
    }
    b1[g] = (nn < HID) ? (bih1[g * HID + nn] + bhh1[g * HID + nn]) : 0.0f;
    b2[g] = (nn < HID) ? (bih2[g * HID + nn] + bhh2[g * HID + nn]) : 0.0f;
  }
  #pragma unroll
  for (int kk = 0; kk < 3; ++kk)
    #pragma unroll
    for (int j = 0; j < 2; ++j) {
      int k = 4 * kk + 2 * kh + j;
      wl[kk][j] = (nn < OUTN && k < HID) ? Wl[nn * HID + k] : 0.0f;
    }
  blv = (nn < OUTN) ? bl[nn] : 0.0f;

  // ---------------- zero recurrent state ----------------
  #pragma unroll
  for (int r = 0; r < 8; ++r) {
    ldsH1[(kh * 8 + r) * PITCH + nn] = 0.0f;
    ldsH2[(kh * 8 + r) * PITCH + nn] = 0.0f;
  }
  lds_fence();

  v8f c1 = {}, c2 = {};
  const float* xrow  = x  + (size_t)(b0 + nn) * (SEQ * TIN);
  const float* tarow = ta + (size_t)(b0 + nn) * (SEQ * 4);

  // One full stacked-cell step given the layer-1 A operand (6 K-steps of v2f).
  auto cellStep = [&](const v2f* a1) {
    // ----- layer 1 gates -----
    v8f g4[4];
    #pragma unroll
    for (int g = 0; g < 4; ++g) {
      v8f acc;
      #pragma unroll
      for (int r = 0; r < 8; ++r) acc[r] = b1[g];
      #pragma unroll
      for (int kk = 0; kk < 6; ++kk) acc = WMMA_F32(a1[kk], w1[g][kk], acc);
      g4[g] = acc;
    }
    v8f h1d;
    #pragma unroll
    for (int r = 0; r < 8; ++r) {
      float iv = sigf(g4[0][r]);
      float fv = sigf(g4[1][r]);
      float gv = tanhfast(g4[2][r]);
      float ov = sigf(g4[3][r]);
      float c  = fv * c1[r] + iv * gv;
      c1[r]    = c;
      h1d[r]   = ov * tanhfast(c);
    }
    #pragma unroll
    for (int r = 0; r < 8; ++r) ldsH1[(kh * 8 + r) * PITCH + nn] = h1d[r];
    lds_fence();

    // ----- layer 2 A from [h1 | h2] -----
    v2f a2[5];
    #pragma unroll
    for (int kk = 0; kk < 5; ++kk)
      #pragma unroll
      for (int j = 0; j < 2; ++j) {
        int k = 4 * kk + 2 * kh + j;
        float v = 0.0f;
        if (k < HID)          v = ldsH1[nn * PITCH + k];
        else if (k < 2 * HID) v = ldsH2[nn * PITCH + (k - HID)];
        a2[kk][j] = v;
      }

    #pragma unroll
    for (int g = 0; g < 4; ++g) {
      v8f acc;
      #pragma unroll
      for (int r = 0; r < 8; ++r) acc[r] = b2[g];
      #pragma unroll
      for (int kk = 0; kk < 5; ++kk) acc = WMMA_F32(a2[kk], w2[g][kk], acc);
      g4[g] = acc;
    }
    v8f h2d;
    #pragma unroll
    for (int r = 0; r < 8; ++r) {
      float iv = sigf(g4[0][r]);
      float fv = sigf(g4[1][r]);
      float gv = tanhfast(g4[2][r]);
      float ov = sigf(g4[3][r]);
      float c  = fv * c2[r] + iv * gv;
      c2[r]    = c;
      h2d[r]   = ov * tanhfast(c);
    }
    #pragma unroll
    for (int r = 0; r < 8; ++r) ldsH2[(kh * 8 + r) * PITCH + nn] = h2d[r];
    lds_fence();
  };

  // Output projection: out = h2 @ Wl^T + bl (D-layout result).
  auto projOut = [&]() -> v8f {
    v2f aa[3];
    #pragma unroll
    for (int kk = 0; kk < 3; ++kk)
      #pragma unroll
      for (int j = 0; j < 2; ++j) {
        int k = 4 * kk + 2 * kh + j;
        aa[kk][j] = (k < HID) ? ldsH2[nn * PITCH + k] : 0.0f;
      }
    v8f acc;
    #pragma unroll
    for (int r = 0; r < 8; ++r) acc[r] = blv;
    #pragma unroll
    for (int kk = 0; kk < 3; ++kk) acc = WMMA_F32(aa[kk], wl[kk], acc);
    return acc;
  };

  // ---------------- encoder ----------------
  for (int t = 0; t < SEQ; ++t) {
    v2f a1[6];
    #pragma unroll
    for (int kk = 0; kk < 6; ++kk)
      #pragma unroll
      for (int j = 0; j < 2; ++j) {
        int k = 4 * kk + 2 * kh + j;
        float v = 0.0f;
        if (k < TIN)            v = xrow[t * TIN + k];
        else if (k < TIN + HID) v = ldsH1[nn * PITCH + (k - TIN)];
        a1[kk][j] = v;
      }
    cellStep(a1);
  }

  // seed decoder "out" state
  {
    v8f od = projOut();
    #pragma unroll
    for (int r = 0; r < 8; ++r) ldsOut[(kh * 8 + r) * PITCH + nn] = od[r];
    lds_fence();
  }

  // ---------------- decoder ----------------
  for (int t = 0; t < SEQ; ++t) {
    v2f a1[6];
    #pragma unroll
    for (int kk = 0; kk < 6; ++kk)
      #pragma unroll
      for (int j = 0; j < 2; ++j) {
        int k = 4 * kk + 2 * kh + j;
        float v = 0.0f;
        if (k < 4)              v = tarow[t * 4 + k];
        else if (k < TIN)       v = ldsOut[nn * PITCH + (k - 4)];
        else if (k < TIN + HID) v = ldsH1[nn * PITCH + (k - TIN)];
        a1[kk][j] = v;
      }
    cellStep(a1);

    v8f od = projOut();
    if (nn < OUTN) {
      #pragma unroll
      for (int r = 0; r < 8; ++r)
        outp[((size_t)(b0 + kh * 8 + r) * SEQ + t) * OUTN + nn] = od[r];
    }
    #pragma unroll
    for (int r = 0; r < 8; ++r) ldsOut[(kh * 8 + r) * PITCH + nn] = od[r];
    lds_fence();
  }
}

extern "C" void kernel_launch(void* const* d_in, const int* in_sizes, int n_in,
                              void* d_out, int out_size, void* d_ws, size_t ws_size,
                              hipStream_t stream) {
  (void)n_in; (void)d_ws; (void)ws_size; (void)out_size;
  const float* x    = (const float*)d_in[0];
  const float* ta   = (const float*)d_in[1];
  const float* Wih1 = (const float*)d_in[2];
  const float* Whh1 = (const float*)d_in[3];
  const float* bih1 = (const float*)d_in[4];
  const float* bhh1 = (const float*)d_in[5];
  const float* Wih2 = (const float*)d_in[6];
  const float* Whh2 = (const float*)d_in[7];
  const float* bih2 = (const float*)d_in[8];
  const float* bhh2 = (const float*)d_in[9];
  const float* Wl   = (const float*)d_in[10];
  const float* bl   = (const float*)d_in[11];
  float* out = (float*)d_out;

  const int Btot = in_sizes[0] / (SEQ * TIN);      // 32768
  const int blocks = Btot / 128;                   // 8 waves x 16 batch per block
  lstm_encdec_kernel<<<blocks, 256, 0, stream>>>(
      x, ta, Wih1, Whh1, bih1, bhh1, Wih2, Whh2, bih2, bhh2, Wl, bl, out);
}